// PositionEmbeding_88691074663153
// MI455X (gfx1250) — compile-verified
//
#include <hip/hip_runtime.h>
#include <hip/hip_bf16.h>

// out[b, s, d] = x[b, s, d] + pe[s, d]
//   pe[s, d] = sin((s+1) * 10000^(-d/D))       for even d
//            = cos((s+1) * 10000^(-(d+1)/D))   for odd  d
//
// Shapes fixed by the reference: B=4, S=4096, D=1024, fp32.
// Memory-bound: 128 MiB traffic -> ~5.5 us floor at 23.3 TB/s.
// Strategy: one block per seq row s (256 threads x float4 = 1024 = D),
// each thread computes its 4 pe values once and streams all 4 batches
// with nontemporal B128 loads/stores.

typedef __attribute__((ext_vector_type(4))) float v4f;

__global__ __launch_bounds__(256) void pe_add_kernel(const float* __restrict__ x,
                                                     float* __restrict__ out) {
  constexpr int S = 4096;
  constexpr int D = 1024;
  constexpr int B = 4;

  const int s  = blockIdx.x;        // 0..4095  (one seq row per block)
  const int d0 = threadIdx.x << 2;  // 0..1020  (float4 chunk of the row)

  const float p = (float)(s + 1);
  // -ln(10000)/D
  constexpr float kC = -9.210340371976184f / (float)D;

  float pe0, pe1, pe2, pe3;
  {
    // even d -> exponent index d, sin; odd d -> exponent index d+1, cos
    // d0 is a multiple of 4, so lanes are (even, odd, even, odd).
    const float e0 = (float)(d0 + 0);
    const float e1 = (float)(d0 + 2);
    const float e2 = (float)(d0 + 2);
    const float e3 = (float)(d0 + 4);
    const float a0 = p * __expf(e0 * kC);
    const float a1 = p * __expf(e1 * kC);
    const float a2 = p * __expf(e2 * kC);
    const float a3 = p * __expf(e3 * kC);
    pe0 = __sinf(a0);
    pe1 = __cosf(a1);
    pe2 = __sinf(a2);
    pe3 = __cosf(a3);
  }
  const v4f pev = {pe0, pe1, pe2, pe3};

  // Apply the same pe chunk to all 4 batches (pe amortized 4x).
  const size_t row = (size_t)s * D + (size_t)d0;
  constexpr size_t batch_stride = (size_t)S * D;

#pragma unroll
  for (int b = 0; b < B; ++b) {
    const size_t idx = (size_t)b * batch_stride + row;
    const v4f* __restrict__ xp = (const v4f*)(x + idx);
    v4f v = __builtin_nontemporal_load(xp);   // TH=NT: zero-reuse stream
    v += pev;
    __builtin_nontemporal_store(v, (v4f*)(out + idx));
  }
}

extern "C" void kernel_launch(void* const* d_in, const int* in_sizes, int n_in,
                              void* d_out, int out_size, void* d_ws, size_t ws_size,
                              hipStream_t stream) {
  (void)in_sizes; (void)n_in; (void)d_ws; (void)ws_size; (void)out_size;

  const float* x = (const float*)d_in[0];
  float* out     = (float*)d_out;

  constexpr int S = 4096;
  dim3 grid(S);     // one block per seq row
  dim3 block(256);  // 256 threads * float4 = 1024 = D; 8 waves (wave32)

  pe_add_kernel<<<grid, block, 0, stream>>>(x, out);
}